// TransR_39316130628051
// MI455X (gfx1250) — compile-verified
//
#include <hip/hip_runtime.h>
#include <hip/hip_bf16.h>

#define DIM     128
#define BATCH   8192
#define MARGIN  1.0f

typedef float v2f __attribute__((ext_vector_type(2)));
typedef float v8f __attribute__((ext_vector_type(8)));

// One wave32 per triple. Computes score(b) = sum |norm(h*Mr) + norm(r) - norm(t*Mr)|
// using V_WMMA_F32_16X16X4_F32: A rows 0/1 carry h/t, B tiles stream Mr from L2.
__global__ __launch_bounds__(256) void transr_score_kernel(
    const int*   __restrict__ pos_triples,
    const int*   __restrict__ neg_triples,
    const float* __restrict__ entities,
    const float* __restrict__ relations,
    const float* __restrict__ transfer,
    float*       __restrict__ scores)   // [2*BATCH]: pos scores then neg scores
{
    __shared__ float sh[8][2][DIM];      // per-wave staging of h and t vectors

    const int lane = threadIdx.x & 31;
    const int wib  = threadIdx.x >> 5;               // wave index in block (0..7)
    const int w    = blockIdx.x * 8 + wib;           // 0..16383
    const int b    = (w < BATCH) ? w : (w - BATCH);
    const int* __restrict__ trip = (w < BATCH) ? pos_triples : neg_triples;

    const int head = trip[3 * b + 0];
    const int rel  = trip[3 * b + 1];
    const int tail = trip[3 * b + 2];

    const float* __restrict__ hvec = entities  + (size_t)head * DIM;
    const float* __restrict__ tvec = entities  + (size_t)tail * DIM;
    const float* __restrict__ rvec = relations + (size_t)rel  * DIM;
    const float* __restrict__ Mr   = transfer  + (size_t)rel  * (DIM * DIM);

    // Stage h and t into LDS so every lane can fetch its A-fragment slots.
    for (int i = lane; i < DIM; i += 32) {
        sh[wib][0][i] = hvec[i];
        sh[wib][1][i] = tvec[i];
    }
    __syncthreads();

    const int M     = lane & 15;    // row of A / column of B owned by this lane
    const int khalf = lane >> 4;    // 0 -> K{0,1}, 1 -> K{2,3}

    // Branchless A-row source: row 0 -> h, row 1 -> t, rows >=2 alias h but are
    // zeroed by 'active' selects (data is finite, so the select is exact).
    const float* __restrict__ aptr = sh[wib][0] + ((M == 1) ? DIM : 0);
    const bool   active = (M < 2);

    v8f acc[8];
#pragma unroll
    for (int e = 0; e < 8; ++e) acc[e] = (v8f){0.f,0.f,0.f,0.f,0.f,0.f,0.f,0.f};

    // D(16x16) += A(16x4) * B(4x16), 32 K-steps x 8 N-tiles = 256 f32 WMMAs.
    for (int k4 = 0; k4 < 32; ++k4) {
        const int kbase = k4 * 4 + 2 * khalf;      // even -> 8B-aligned LDS load

        v2f av = *(const v2f*)(aptr + kbase);      // one ds_load_b64, no branches
        v2f a;
        a.x = active ? av.x : 0.f;
        a.y = active ? av.y : 0.f;

        // B fragment base: rows kbase / kbase+1 of Mr, column = N-tile*16 + M.
        const float* __restrict__ Brow = Mr + (size_t)kbase * DIM + M;
#pragma unroll
        for (int e = 0; e < 8; ++e) {
            v2f bf;
            bf.x = Brow[e * 16];
            bf.y = Brow[e * 16 + DIM];
            acc[e] = __builtin_amdgcn_wmma_f32_16x16x4_f32(
                false, a, false, bf, (short)0, acc[e], false, false);
        }
    }

    // Extract projections: D VGPR0 lanes 0..15 = row M=0 (h'), VGPR1 = row M=1 (t').
    // Lanes 16..31 hold rows 8/9 which are exactly zero -> safe in reductions.
    float hp[8], tp[8], rr[8];
    float hn = 0.f, tn = 0.f, rn = 0.f;
#pragma unroll
    for (int e = 0; e < 8; ++e) {
        hp[e] = acc[e][0];
        tp[e] = acc[e][1];
        rr[e] = (lane < 16) ? rvec[e * 16 + lane] : 0.f;
        hn += hp[e] * hp[e];
        tn += tp[e] * tp[e];
        rn += rr[e] * rr[e];
    }
#pragma unroll
    for (int off = 16; off >= 1; off >>= 1) {
        hn += __shfl_xor(hn, off, 32);
        tn += __shfl_xor(tn, off, 32);
        rn += __shfl_xor(rn, off, 32);
    }

    const float eps = 1e-12f;
    const float ih = 1.0f / fmaxf(sqrtf(hn), eps);
    const float it = 1.0f / fmaxf(sqrtf(tn), eps);
    const float ir = 1.0f / fmaxf(sqrtf(rn), eps);

    float s = 0.f;
#pragma unroll
    for (int e = 0; e < 8; ++e)
        s += fabsf(hp[e] * ih + rr[e] * ir - tp[e] * it);
#pragma unroll
    for (int off = 16; off >= 1; off >>= 1)
        s += __shfl_xor(s, off, 32);

    if (lane == 0) scores[w] = s;
}

// Deterministic single-block reduction: mean(relu(pos - neg + MARGIN)).
__global__ __launch_bounds__(256) void transr_loss_kernel(
    const float* __restrict__ scores, float* __restrict__ out)
{
    __shared__ float red[256];
    float local = 0.f;
    for (int b = threadIdx.x; b < BATCH; b += 256) {
        float v = scores[b] - scores[b + BATCH] + MARGIN;
        local += (v > 0.f) ? v : 0.f;
    }
    red[threadIdx.x] = local;
    __syncthreads();
    for (int s2 = 128; s2 > 0; s2 >>= 1) {
        if ((int)threadIdx.x < s2) red[threadIdx.x] += red[threadIdx.x + s2];
        __syncthreads();
    }
    if (threadIdx.x == 0) out[0] = red[0] / (float)BATCH;
}

extern "C" void kernel_launch(void* const* d_in, const int* in_sizes, int n_in,
                              void* d_out, int out_size, void* d_ws, size_t ws_size,
                              hipStream_t stream)
{
    // setup_inputs order: positive_triples, negative_triples, entities,
    //                     relations, transfer_matrix
    const int*   pos = (const int*)  d_in[0];   // JAX x64 disabled -> int32
    const int*   neg = (const int*)  d_in[1];
    const float* ent = (const float*)d_in[2];
    const float* rel = (const float*)d_in[3];
    const float* tm  = (const float*)d_in[4];

    float* scores = (float*)d_ws;               // 2*BATCH floats = 64 KB

    transr_score_kernel<<<(2 * BATCH) / 8, 256, 0, stream>>>(
        pos, neg, ent, rel, tm, scores);
    transr_loss_kernel<<<1, 256, 0, stream>>>(scores, (float*)d_out);
}